// RGAT_64450279244474
// MI455X (gfx1250) — compile-verified
//
#include <hip/hip_runtime.h>
#include <hip/hip_bf16.h>

#define NN 100000
#define HIDD 128
#define HH 2
#define DD 64
#define RRR 3
#define LLL 2
#define EE 500000
#define CCC 64

typedef __attribute__((ext_vector_type(2))) float v2f;
typedef __attribute__((ext_vector_type(8))) float v8f;

// Order-preserving float->uint encoding for atomicMax-based segment_max.
__device__ __forceinline__ unsigned fenc(float f) {
  unsigned u = __float_as_uint(f);
  return (u & 0x80000000u) ? ~u : (u | 0x80000000u);
}
__device__ __forceinline__ float fdec(unsigned k) {
  unsigned u = (k & 0x80000000u) ? (k ^ 0x80000000u) : ~k;
  return __uint_as_float(u);
}

// out[N,128] = A[N,128] @ W[128,128] in exact fp32 via V_WMMA_F32_16X16X4_F32.
// Block = 256 threads = 8 waves = 8 column tiles of 16; each block covers 64 rows
// (4 row-tiles per wave) so each B register pair feeds 4 WMMAs (4x less B traffic).
__global__ __launch_bounds__(256) void k_gemm128(const float* __restrict__ A,
                                                 const float* __restrict__ W,
                                                 float* __restrict__ out) {
  __shared__ float As[64][132];  // pad: bank = (4r + c) mod 64, conflict-free per tile
  const int lane = threadIdx.x & 31;
  const int wv = threadIdx.x >> 5;
  const int row0 = blockIdx.x << 6;
  for (int i = threadIdx.x; i < 64 * 128; i += 256) {
    int r = i >> 7, c = i & 127;
    int gr = row0 + r;
    As[r][c] = (gr < NN) ? A[(size_t)gr * 128 + c] : 0.f;
  }
  __syncthreads();
  const int m = lane & 15;
  const int hf = lane >> 4;  // 0: K pair 0/1, 1: K pair 2/3
  const int n0 = wv << 4;
  v8f acc0 = {}, acc1 = {}, acc2 = {}, acc3 = {};
#pragma unroll
  for (int k0 = 0; k0 < 128; k0 += 4) {
    const int ka = k0 + 2 * hf;
    v2f b;
    b.x = W[ka * 128 + n0 + m];
    b.y = W[(ka + 1) * 128 + n0 + m];
    v2f a;
    a.x = As[m][ka];      a.y = As[m][ka + 1];
    acc0 = __builtin_amdgcn_wmma_f32_16x16x4_f32(false, a, false, b, (short)0, acc0, false, false);
    a.x = As[16 + m][ka]; a.y = As[16 + m][ka + 1];
    acc1 = __builtin_amdgcn_wmma_f32_16x16x4_f32(false, a, false, b, (short)0, acc1, false, false);
    a.x = As[32 + m][ka]; a.y = As[32 + m][ka + 1];
    acc2 = __builtin_amdgcn_wmma_f32_16x16x4_f32(false, a, false, b, (short)0, acc2, false, false);
    a.x = As[48 + m][ka]; a.y = As[48 + m][ka + 1];
    acc3 = __builtin_amdgcn_wmma_f32_16x16x4_f32(false, a, false, b, (short)0, acc3, false, false);
  }
#pragma unroll
  for (int v = 0; v < 8; ++v) {
    int gr = row0 + v + 8 * hf;
    if (gr < NN)      out[(size_t)gr * 128 + n0 + m] = acc0[v];
    if (gr + 16 < NN) out[(size_t)(gr + 16) * 128 + n0 + m] = acc1[v];
    if (gr + 32 < NN) out[(size_t)(gr + 32) * 128 + n0 + m] = acc2[v];
    if (gr + 48 < NN) out[(size_t)(gr + 48) * 128 + n0 + m] = acc3[v];
  }
}

// Per-node attention logits: el[n,h] = <feat[n,h,:], al[h,:]>, er likewise.
__global__ __launch_bounds__(64) void k_dots(const float* __restrict__ F,
                                             const float* __restrict__ al,
                                             const float* __restrict__ ar,
                                             float* __restrict__ el,
                                             float* __restrict__ er) {
  const int n = blockIdx.x;
  const int head = threadIdx.x >> 5;
  const int lane = threadIdx.x & 31;
  const float* f = F + (size_t)n * 128 + head * 64;
  const float* a = al + head * 64;
  const float* b = ar + head * 64;
  float f0 = f[lane], f1 = f[lane + 32];
  float pl = f0 * a[lane] + f1 * a[lane + 32];
  float pr = f0 * b[lane] + f1 * b[lane + 32];
  for (int o = 16; o; o >>= 1) {
    pl += __shfl_xor(pl, o, 32);
    pr += __shfl_xor(pr, o, 32);
  }
  if (lane == 0) {
    el[n * 2 + head] = pl;
    er[n * 2 + head] = pr;
  }
}

__global__ void k_init_seg(unsigned* __restrict__ menc, float* __restrict__ ssum, int n) {
  int i = blockIdx.x * blockDim.x + threadIdx.x;
  if (i < n) { menc[i] = 0u; ssum[i] = 0.f; }
}

// acc starts at sum over relations of bias (each relation adds its own bias).
__global__ void k_init_acc(float* __restrict__ acc, const float* __restrict__ bsl, int n) {
  int i = blockIdx.x * blockDim.x + threadIdx.x;
  if (i < n) {
    int c = i & 127;
    acc[i] = bsl[c] + bsl[128 + c] + bsl[256 + c];
  }
}

// Pass 1: e = leaky_relu(el[src]+er[dst]); segment max over dst via atomicMax.
__global__ __launch_bounds__(256) void k_edge1(const int* __restrict__ src, const int* __restrict__ dst,
                                               const float* __restrict__ el, const float* __restrict__ er,
                                               float* __restrict__ ebuf, unsigned* __restrict__ menc) {
  int gid = blockIdx.x * 256 + threadIdx.x;
  if (gid >= EE * HH) return;
  int e = gid >> 1, h = gid & 1;
  int s = src[e], d = dst[e];
  float v = el[s * 2 + h] + er[d * 2 + h];
  v = (v > 0.f) ? v : 0.2f * v;
  ebuf[gid] = v;
  atomicMax(&menc[d * 2 + h], fenc(v));
}

// Pass 2: ex = exp(e - m[dst]); segment sum over dst.
__global__ __launch_bounds__(256) void k_edge2(const int* __restrict__ dst,
                                               float* __restrict__ ebuf, const unsigned* __restrict__ menc,
                                               float* __restrict__ ssum) {
  int gid = blockIdx.x * 256 + threadIdx.x;
  if (gid >= EE * HH) return;
  int e = gid >> 1, h = gid & 1;
  int d = dst[e];
  float ex = expf(ebuf[gid] - fdec(menc[d * 2 + h]));
  ebuf[gid] = ex;
  atomicAdd(&ssum[d * 2 + h], ex);
}

// Pass 3: alpha = ex / ssum[dst]  (in place; pass 4 reads one value per edge-head)
__global__ __launch_bounds__(256) void k_edge_alpha(const int* __restrict__ dst,
                                                    float* __restrict__ ebuf,
                                                    const float* __restrict__ ssum) {
  int gid = blockIdx.x * 256 + threadIdx.x;
  if (gid >= EE * HH) return;
  int e = gid >> 1, h = gid & 1;
  ebuf[gid] = ebuf[gid] / ssum[dst[e] * 2 + h];
}

// Pass 4: acc[dst] += alpha * feat[src]  (one thread per edge-channel; L2-resident atomics)
__global__ __launch_bounds__(256) void k_edge3(const int* __restrict__ src, const int* __restrict__ dst,
                                               const float* __restrict__ F, const float* __restrict__ alpha,
                                               float* __restrict__ acc) {
  long long gid = (long long)blockIdx.x * 256 + threadIdx.x;
  if (gid >= (long long)EE * 128) return;
  int e = (int)(gid >> 7), c = (int)(gid & 127);
  int h = c >> 6;
  int s = src[e], d = dst[e];
  atomicAdd(&acc[(size_t)d * 128 + c], F[(size_t)s * 128 + c] * alpha[e * 2 + h]);
}

__global__ void k_relu(const float* __restrict__ acc, float* __restrict__ out, int n) {
  int i = blockIdx.x * blockDim.x + threadIdx.x;
  if (i < n) out[i] = fmaxf(acc[i], 0.f);
}

__global__ __launch_bounds__(128) void k_zero128(float* a, float* b) {
  a[threadIdx.x] = 0.f;
  b[threadIdx.x] = 0.f;
}

// Column sums / sums-of-squares for BatchNorm (biased variance).
__global__ __launch_bounds__(128) void k_bnstats(const float* __restrict__ Hm,
                                                 float* __restrict__ cs, float* __restrict__ cq) {
  int c = threadIdx.x;
  int r0 = blockIdx.x * 256;
  int rend = r0 + 256;
  if (rend > NN) rend = NN;
  float s = 0.f, q = 0.f;
  for (int r = r0; r < rend; ++r) {
    float v = Hm[(size_t)r * 128 + c];
    s += v;
    q += v * v;
  }
  atomicAdd(&cs[c], s);
  atomicAdd(&cq[c], q);
}

__global__ __launch_bounds__(128) void k_bnfin(const float* __restrict__ cs, const float* __restrict__ cq,
                                               const float* __restrict__ bn_w, const float* __restrict__ bn_b,
                                               float* __restrict__ scale, float* __restrict__ shift) {
  int c = threadIdx.x;
  float mu = cs[c] * (1.0f / NN);
  float var = cq[c] * (1.0f / NN) - mu * mu;
  float inv = rsqrtf(var + 1e-5f);
  float sc = bn_w[c] * inv;
  scale[c] = sc;
  shift[c] = bn_b[c] - mu * sc;
}

// Classifier: out[N,64] = BN(h) @ clf_W + clf_b, BN fused into A-tile load.
// Block = 128 threads = 4 waves = 4 col tiles (C=64); 64 rows per block, 4 WMMAs per B pair.
__global__ __launch_bounds__(128) void k_clf(const float* __restrict__ Hm,
                                             const float* __restrict__ scale, const float* __restrict__ shift,
                                             const float* __restrict__ Wc, const float* __restrict__ bc,
                                             float* __restrict__ out) {
  __shared__ float As[64][132];
  const int lane = threadIdx.x & 31;
  const int wv = threadIdx.x >> 5;
  const int row0 = blockIdx.x << 6;
  for (int i = threadIdx.x; i < 64 * 128; i += 128) {
    int r = i >> 7, c = i & 127;
    int gr = row0 + r;
    As[r][c] = (gr < NN) ? Hm[(size_t)gr * 128 + c] * scale[c] + shift[c] : 0.f;
  }
  __syncthreads();
  const int m = lane & 15;
  const int hf = lane >> 4;
  const int n0 = wv << 4;
  v8f acc0 = {}, acc1 = {}, acc2 = {}, acc3 = {};
#pragma unroll
  for (int k0 = 0; k0 < 128; k0 += 4) {
    const int ka = k0 + 2 * hf;
    v2f b;
    b.x = Wc[ka * 64 + n0 + m];
    b.y = Wc[(ka + 1) * 64 + n0 + m];
    v2f a;
    a.x = As[m][ka];      a.y = As[m][ka + 1];
    acc0 = __builtin_amdgcn_wmma_f32_16x16x4_f32(false, a, false, b, (short)0, acc0, false, false);
    a.x = As[16 + m][ka]; a.y = As[16 + m][ka + 1];
    acc1 = __builtin_amdgcn_wmma_f32_16x16x4_f32(false, a, false, b, (short)0, acc1, false, false);
    a.x = As[32 + m][ka]; a.y = As[32 + m][ka + 1];
    acc2 = __builtin_amdgcn_wmma_f32_16x16x4_f32(false, a, false, b, (short)0, acc2, false, false);
    a.x = As[48 + m][ka]; a.y = As[48 + m][ka + 1];
    acc3 = __builtin_amdgcn_wmma_f32_16x16x4_f32(false, a, false, b, (short)0, acc3, false, false);
  }
  float bb = bc[n0 + m];
#pragma unroll
  for (int v = 0; v < 8; ++v) {
    int gr = row0 + v + 8 * hf;
    if (gr < NN)      out[(size_t)gr * 64 + n0 + m] = acc0[v] + bb;
    if (gr + 16 < NN) out[(size_t)(gr + 16) * 64 + n0 + m] = acc1[v] + bb;
    if (gr + 32 < NN) out[(size_t)(gr + 32) * 64 + n0 + m] = acc2[v] + bb;
    if (gr + 48 < NN) out[(size_t)(gr + 48) * 64 + n0 + m] = acc3[v] + bb;
  }
}

extern "C" void kernel_launch(void* const* d_in, const int* in_sizes, int n_in,
                              void* d_out, int out_size, void* d_ws, size_t ws_size,
                              hipStream_t stream) {
  const float* feat = (const float*)d_in[0];
  const float* Ws   = (const float*)d_in[1];
  const float* als  = (const float*)d_in[2];
  const float* ars  = (const float*)d_in[3];
  const float* bs   = (const float*)d_in[4];
  const float* bn_w = (const float*)d_in[5];
  const float* bn_b = (const float*)d_in[6];
  const float* clfW = (const float*)d_in[7];
  const float* clfb = (const float*)d_in[8];
  const int* srcA   = (const int*)d_in[9];
  const int* dstA   = (const int*)d_in[10];
  (void)in_sizes; (void)n_in; (void)out_size; (void)ws_size;

  char* wsp = (char*)d_ws;
  size_t off = 0;
  auto alloc = [&](size_t bytes) -> void* {
    void* p = wsp + off;
    off += (bytes + 255) & ~(size_t)255;
    return p;
  };
  float* hbuf = (float*)alloc((size_t)NN * 128 * 4);  // layer-1 output / layer-2 input
  float* fbuf = (float*)alloc((size_t)NN * 128 * 4);  // per-relation Wh ; final h2
  float* acc  = (float*)alloc((size_t)NN * 128 * 4);  // relation-summed aggregation
  float* elb  = (float*)alloc((size_t)NN * 2 * 4);
  float* erb  = (float*)alloc((size_t)NN * 2 * 4);
  float* ebuf = (float*)alloc((size_t)EE * 2 * 4);    // e -> ex -> alpha (in place)
  unsigned* menc = (unsigned*)alloc((size_t)NN * 2 * 4);
  float* ssum = (float*)alloc((size_t)NN * 2 * 4);
  float* cs = (float*)alloc(128 * 4);
  float* cq = (float*)alloc(128 * 4);
  float* scale = (float*)alloc(128 * 4);
  float* shift = (float*)alloc(128 * 4);

  const int ROWB = (NN + 63) / 64;  // 1563 blocks of 64 rows
  const int NH = NN * 2;
  const int EH = EE * 2;
  const int ecBlocks = (int)(((long long)EE * 128 + 255) / 256);

  for (int l = 0; l < LLL; ++l) {
    const float* hin = (l == 0) ? feat : hbuf;
    float* hout = (l == 0) ? hbuf : fbuf;  // layer2 output reuses fbuf after last relation
    k_init_acc<<<(NN * 128 + 255) / 256, 256, 0, stream>>>(acc, bs + (size_t)l * RRR * 128, NN * 128);
    for (int r = 0; r < RRR; ++r) {
      const float* Wlr = Ws + ((size_t)l * RRR + r) * 128 * 128;
      const float* al  = als + ((size_t)l * RRR + r) * 128;
      const float* ar  = ars + ((size_t)l * RRR + r) * 128;
      const int* src   = srcA + (size_t)r * EE;
      const int* dst   = dstA + (size_t)r * EE;
      k_gemm128<<<ROWB, 256, 0, stream>>>(hin, Wlr, fbuf);
      k_dots<<<NN, 64, 0, stream>>>(fbuf, al, ar, elb, erb);
      k_init_seg<<<(NH + 255) / 256, 256, 0, stream>>>(menc, ssum, NH);
      k_edge1<<<(EH + 255) / 256, 256, 0, stream>>>(src, dst, elb, erb, ebuf, menc);
      k_edge2<<<(EH + 255) / 256, 256, 0, stream>>>(dst, ebuf, menc, ssum);
      k_edge_alpha<<<(EH + 255) / 256, 256, 0, stream>>>(dst, ebuf, ssum);
      k_edge3<<<ecBlocks, 256, 0, stream>>>(src, dst, fbuf, ebuf, acc);
    }
    k_relu<<<(NN * 128 + 255) / 256, 256, 0, stream>>>(acc, hout, NN * 128);
  }
  k_zero128<<<1, 128, 0, stream>>>(cs, cq);
  k_bnstats<<<(NN + 255) / 256, 128, 0, stream>>>(fbuf, cs, cq);
  k_bnfin<<<1, 128, 0, stream>>>(cs, cq, bn_w, bn_b, scale, shift);
  k_clf<<<ROWB, 128, 0, stream>>>(fbuf, scale, shift, clfW, clfb, (float*)d_out);
}